// PseudoLabelGenerator_53678501266104
// MI455X (gfx1250) — compile-verified
//
#include <hip/hip_runtime.h>

#define CONF_THR 1e-4f
#define NMS_THR  0.5f

typedef unsigned int u32x4 __attribute__((ext_vector_type(4)));
typedef int          i32x4 __attribute__((ext_vector_type(4)));
typedef int          i32x8 __attribute__((ext_vector_type(8)));
typedef float        v2f   __attribute__((ext_vector_type(2)));
typedef float        v8f   __attribute__((ext_vector_type(8)));

#if defined(__has_builtin)
#if __has_builtin(__builtin_amdgcn_tensor_load_to_lds) && __has_builtin(__builtin_amdgcn_s_wait_tensorcnt)
#define HAVE_TDM 1
#endif
#if __has_builtin(__builtin_amdgcn_wmma_f32_16x16x4_f32)
#define HAVE_WMMA_RED 1
#endif
#endif

#if defined(HAVE_TDM)
#pragma message("CDNA5 PATH: tensor_load_to_lds builtin ENABLED")
#else
#pragma message("CDNA5 PATH: tensor_load_to_lds builtin MISSING -> float4 fallback")
#endif
#if defined(HAVE_WMMA_RED)
#pragma message("CDNA5 PATH: wmma_f32_16x16x4_f32 reduction ENABLED")
#else
#pragma message("CDNA5 PATH: wmma_f32_16x16x4_f32 MISSING -> LDS tree fallback")
#endif

// ---------------------------------------------------------------------------
// Exact f32 32-lane sum via two V_WMMA_F32_16X16X4_F32 (see analysis).
// Requires EXEC all-ones. Returns the wave sum broadcast to every lane.
// ---------------------------------------------------------------------------
#if defined(HAVE_WMMA_RED)
__device__ __forceinline__ float wave_sum_wmma(float p) {
    v2f a;   a[0] = p;    a[1] = p;
    v2f one; one[0] = 1.0f; one[1] = 1.0f;
    v8f c = {};
    v8f d = __builtin_amdgcn_wmma_f32_16x16x4_f32(false, a, false, one, (short)0, c,
                                                  false, false);
    float s = (d[0] + d[1]) + (d[2] + d[3]) + ((d[4] + d[5]) + (d[6] + d[7]));
    v2f a2; a2[0] = s; a2[1] = s;
    v8f d2 = __builtin_amdgcn_wmma_f32_16x16x4_f32(false, a2, false, one, (short)0, c,
                                                   false, false);
    return d2[0] * 0.25f;
}

// block (256-thread) sum: 8 wave sums combined through 8 floats of LDS.
// Returns full value on thread 0 (others: partial/meaningless).
__device__ __forceinline__ float block_sum_256(float p, float* lds8) {
    const unsigned t = threadIdx.x;
    float ws = wave_sum_wmma(p);
    if ((t & 31u) == 0u) lds8[t >> 5] = ws;
    __syncthreads();
    float r = 0.0f;
    if (t == 0) {
        r = ((lds8[0] + lds8[1]) + (lds8[2] + lds8[3])) +
            ((lds8[4] + lds8[5]) + (lds8[6] + lds8[7]));
    }
    __syncthreads();
    return r;
}
#endif

// ---------------------------------------------------------------------------
// 0) init: zero keep flags, zero area-max accumulator
// ---------------------------------------------------------------------------
__global__ void init_kernel(float* __restrict__ keepf, unsigned* __restrict__ amaxU, int N) {
    int i = blockIdx.x * blockDim.x + threadIdx.x;
    if (i < N) keepf[i] = 0.0f;
    if (i == 0) *amaxU = 0u;
}

// ---------------------------------------------------------------------------
// 1) Haar edge quality: one 64x64 tile per block, TDM DMA into LDS
// ---------------------------------------------------------------------------
__global__ __launch_bounds__(256) void edge_kernel(const float* __restrict__ feat,
                                                   float* __restrict__ edge, int N) {
    __shared__ float tile[4096];   // 16 KB tile
    __shared__ float red[256];
    const int n = blockIdx.x;
    const unsigned t = threadIdx.x;
    const float* src = feat + (size_t)n * 4096;

#if defined(HAVE_TDM)
    if (t < 32) {   // wave 0 only, EXEC all-ones: TDM issues once per wave
        unsigned lds_off = (unsigned)(unsigned long long)(void*)&tile[0];
        unsigned long long ga = (unsigned long long)(const void*)src;
        u32x4 g0;
        g0.x = 1u;                                   // count=1, user mode
        g0.y = lds_off;                              // lds_addr [63:32]
        g0.z = (unsigned)(ga & 0xFFFFFFFFull);       // global_addr lo32
        g0.w = (unsigned)((ga >> 32) & 0x01FFFFFFull) | (2u << 30); // addr hi | type=2
        i32x8 g1;
        g1[0] = (int)(2u << 16);                     // data_size = 4 bytes
        g1[1] = (int)(4096u << 16);                  // tensor_dim0 lo16
        g1[2] = (int)(1u << 16);                     // tensor_dim0 hi | tensor_dim1 = 1
        g1[3] = (int)(4096u << 16);                  // tensor_dim1 hi | tile_dim0 = 4096
        g1[4] = 1;                                   // tile_dim1 = 1, tile_dim2 = 0
        g1[5] = 4096;                                // tensor_dim0_stride lo32
        g1[6] = (int)(4096u << 16);                  // stride0 hi | tensor_dim1_stride lo16
        g1[7] = 0;
        i32x4 z4 = {0, 0, 0, 0};
#if __clang_major__ >= 23
        i32x8 z8 = {0, 0, 0, 0, 0, 0, 0, 0};
        __builtin_amdgcn_tensor_load_to_lds(g0, g1, z4, z4, z8, 0);
#else
        __builtin_amdgcn_tensor_load_to_lds(g0, g1, z4, z4, 0);
#endif
        __builtin_amdgcn_s_wait_tensorcnt(0);
    }
    __syncthreads();
#else
    {   // fallback: cooperative vectorized load
        const float4* s4 = (const float4*)src;
        float4* t4 = (float4*)tile;
        for (int k = t; k < 1024; k += 256) t4[k] = s4[k];
        __syncthreads();
    }
#endif

    float s = 0.0f;
    #pragma unroll
    for (int r = 0; r < 4; ++r) {
        int q  = (int)t + (r << 8);       // quad index 0..1023
        int qy = q >> 5, qx = q & 31;
        const float* p = &tile[(qy << 7) + (qx << 1)];   // (2*qy)*64 + 2*qx
        float a = p[0], b = p[1], c = p[64], d = p[65];
        s += fabsf(a + b - c - d) + fabsf(a - b + c - d) + fabsf(a - b - c + d);
    }

#if defined(HAVE_WMMA_RED)
    float tot = block_sum_256(s, red);
    if (t == 0) edge[n] = tot * (0.5f / 3072.0f);
#else
    red[t] = s;
    __syncthreads();
    for (int st = 128; st > 0; st >>= 1) {
        if (t < (unsigned)st) red[t] += red[t + st];
        __syncthreads();
    }
    if (t == 0) edge[n] = red[0] * (0.5f / 3072.0f);
#endif
}

// ---------------------------------------------------------------------------
// 2) max softmax prob per row: 1 / sum(exp(x - rowmax))
// ---------------------------------------------------------------------------
__global__ __launch_bounds__(256) void maxprob_kernel(const float* __restrict__ logits,
                                                      float* __restrict__ maxprob,
                                                      int N, int C) {
    __shared__ float red[256];
    const int n = blockIdx.x;
    const int t = threadIdx.x;
    const float* row = logits + (size_t)n * C;
    float m = -__builtin_inff();
    for (int c = t; c < C; c += 256) m = fmaxf(m, row[c]);
    red[t] = m;
    __syncthreads();
    for (int st = 128; st > 0; st >>= 1) {
        if (t < st) red[t] = fmaxf(red[t], red[t + st]);
        __syncthreads();
    }
    float rmax = red[0];
    __syncthreads();
    float sum = 0.0f;
    for (int c = t; c < C; c += 256) sum += __expf(row[c] - rmax);
#if defined(HAVE_WMMA_RED)
    float tot = block_sum_256(sum, red);
    if (t == 0) maxprob[n] = 1.0f / tot;
#else
    red[t] = sum;
    __syncthreads();
    for (int st = 128; st > 0; st >>= 1) {
        if (t < st) red[t] += red[t + st];
        __syncthreads();
    }
    if (t == 0) maxprob[n] = 1.0f / red[0];
#endif
}

// ---------------------------------------------------------------------------
// 3) global max(area) via bit-compatible unsigned atomicMax (areas > 0)
// ---------------------------------------------------------------------------
__global__ void area_kernel(const float* __restrict__ boxes, unsigned* __restrict__ amaxU, int N) {
    int i = blockIdx.x * blockDim.x + threadIdx.x;
    if (i >= N) return;
    float w = boxes[i * 4 + 2] - boxes[i * 4 + 0];
    float h = boxes[i * 4 + 3] - boxes[i * 4 + 1];
    atomicMax(amaxU, __float_as_uint(w * h));
}

// ---------------------------------------------------------------------------
// 4) per-box quality
// ---------------------------------------------------------------------------
__global__ void quality_kernel(const float* __restrict__ boxes, const float* __restrict__ scores,
                               const float* __restrict__ edge, const float* __restrict__ maxprob,
                               const unsigned* __restrict__ amaxU,
                               float* __restrict__ quality, int N) {
    int i = blockIdx.x * blockDim.x + threadIdx.x;
    if (i >= N) return;
    float w = boxes[i * 4 + 2] - boxes[i * 4 + 0];
    float h = boxes[i * 4 + 3] - boxes[i * 4 + 1];
    float aspect = fminf(w / h, h / w);
    float area = w * h;
    float amax = __uint_as_float(*amaxU);
    float as = fminf(fmaxf(area / (amax + 1e-6f), 0.0f), 1.0f);
    float det = scores[i] * maxprob[i] * (aspect * as);
    quality[i] = det * 0.4f + 0.3f + edge[i] * 0.3f;
}

// ---------------------------------------------------------------------------
// 5) single-block bitonic argsort on packed 64-bit keys in dynamic LDS
// ---------------------------------------------------------------------------
__global__ __launch_bounds__(1024) void sort_kernel(const float* __restrict__ scores,
                                                    const float* __restrict__ quality,
                                                    int* __restrict__ order, int N, int NP) {
    extern __shared__ unsigned long long keys[];
    const int t = threadIdx.x;
    for (int i = t; i < NP; i += 1024) {
        unsigned long long key;
        if (i < N) {
            float q  = quality[i];
            float ms = (q > CONF_THR) ? scores[i] : -__builtin_inff();
            unsigned u    = __float_as_uint(ms);
            unsigned kasc = (u & 0x80000000u) ? ~u : (u | 0x80000000u);
            key = ((unsigned long long)(~kasc) << 32) | (unsigned)i;
        } else {
            key = ~0ull;
        }
        keys[i] = key;
    }
    __syncthreads();
    for (int k = 2; k <= NP; k <<= 1) {
        for (int j = k >> 1; j > 0; j >>= 1) {
            for (int i = t; i < NP; i += 1024) {
                int ixj = i ^ j;
                if (ixj > i) {
                    bool up = ((i & k) == 0);
                    unsigned long long a = keys[i], b = keys[ixj];
                    if ((a > b) == up) { keys[i] = b; keys[ixj] = a; }
                }
            }
            __syncthreads();
        }
    }
    for (int i = t; i < N; i += 1024)
        order[i] = (int)(unsigned)(keys[i] & 0xFFFFFFFFull);
}

// ---------------------------------------------------------------------------
// 6) gather boxes into sorted order (areas recomputed downstream)
// ---------------------------------------------------------------------------
__global__ void gather_kernel(const float* __restrict__ boxes, const int* __restrict__ order,
                              float* __restrict__ sboxes, int N) {
    int i = blockIdx.x * blockDim.x + threadIdx.x;
    if (i >= N) return;
    int o = order[i];
    const float4* b4 = (const float4*)boxes;
    ((float4*)sboxes)[i] = b4[o];
}

// ---------------------------------------------------------------------------
// 7) pairwise IoU suppression bitmask: block = sorted row i, thread = 32-col word
// ---------------------------------------------------------------------------
__global__ __launch_bounds__(256) void iou_mask_kernel(const float* __restrict__ sboxes,
                                                       unsigned* __restrict__ supmat,
                                                       int N, int NW) {
    const int i = blockIdx.x;
    const int w = threadIdx.x;
    const float4* sb4 = (const float4*)sboxes;
    const float4 bi = sb4[i];                       // uniform -> scalar loads
    const float ai = (bi.z - bi.x) * (bi.w - bi.y);
    unsigned bits = 0u;
    const int j0 = w * 32;
    if (j0 + 31 > i) {
        for (int b = 0; b < 32; ++b) {
            int j = j0 + b;
            if (j <= i || j >= N) continue;
            float4 bj = sb4[j];
            float aj = (bj.z - bj.x) * (bj.w - bj.y);
            float xx1 = fmaxf(bi.x, bj.x);
            float yy1 = fmaxf(bi.y, bj.y);
            float xx2 = fminf(bi.z, bj.z);
            float yy2 = fminf(bi.w, bj.w);
            float inter = fmaxf(xx2 - xx1, 0.0f) * fmaxf(yy2 - yy1, 0.0f);
            float iou = inter / (ai + aj - inter);
            if (iou > NMS_THR) bits |= (1u << b);
        }
    }
    supmat[(size_t)i * NW + w] = bits;
}

// ---------------------------------------------------------------------------
// 8) serial greedy scan: suppressed bitmap in LDS, one row OR per kept box
// ---------------------------------------------------------------------------
__global__ __launch_bounds__(256) void nms_scan_kernel(const float* __restrict__ quality,
                                                       const int* __restrict__ order,
                                                       const unsigned* __restrict__ supmat,
                                                       float* __restrict__ keepf,
                                                       int N, int NW) {
    __shared__ unsigned sup[256];
    __shared__ int cur;
    const int t = threadIdx.x;
    unsigned wbits = 0u;
    for (int b = 0; b < 32; ++b) {
        int j = t * 32 + b;
        if (j >= N) { wbits |= (1u << b); continue; }
        if (!(quality[order[j]] > CONF_THR)) wbits |= (1u << b);
    }
    sup[t] = wbits;
    __syncthreads();

    int i = 0;
    while (true) {
        if (t == 0) {
            int found = -1;
            while (i < N) {
                unsigned m = sup[i >> 5] | ((1u << (i & 31)) - 1u);
                if (m != 0xFFFFFFFFu) { found = (i & ~31) + (__ffs(~m) - 1); break; }
                i = (i & ~31) + 32;
            }
            cur = found;
        }
        __syncthreads();
        int c = cur;
        if (c < 0) break;
        if (t == 0) { keepf[order[c]] = 1.0f; i = c + 1; }
        sup[t] |= supmat[(size_t)c * NW + t];
        __syncthreads();
    }
}

// ---------------------------------------------------------------------------
// 9) assemble outputs: [boxes|score|quality]*keep, keep, labels
// ---------------------------------------------------------------------------
__global__ void finalize_kernel(const float* __restrict__ boxes, const float* __restrict__ scores,
                                const float* __restrict__ quality, const int* __restrict__ labels,
                                const float* __restrict__ keepf,
                                float* __restrict__ out, float* __restrict__ labout, int N) {
    int i = blockIdx.x * blockDim.x + threadIdx.x;
    if (i >= N) return;
    float k = keepf[i];
    out[i * 6 + 0] = boxes[i * 4 + 0] * k;
    out[i * 6 + 1] = boxes[i * 4 + 1] * k;
    out[i * 6 + 2] = boxes[i * 4 + 2] * k;
    out[i * 6 + 3] = boxes[i * 4 + 3] * k;
    out[i * 6 + 4] = scores[i] * k;
    out[i * 6 + 5] = quality[i] * k;
    labout[i] = (float)labels[i];
}

// ---------------------------------------------------------------------------
extern "C" void kernel_launch(void* const* d_in, const int* in_sizes, int n_in,
                              void* d_out, int out_size, void* d_ws, size_t ws_size,
                              hipStream_t stream) {
    const float* boxes  = (const float*)d_in[0];
    const float* scores = (const float*)d_in[1];
    const float* logits = (const float*)d_in[2];
    const float* feat   = (const float*)d_in[3];
    const int*   labels = (const int*)d_in[4];

    const int N  = in_sizes[1];            // 8192
    const int C  = in_sizes[2] / N;        // 1000
    const int NW = (N + 31) / 32;          // 256 suppression words per row
    int NP = 1; while (NP < N) NP <<= 1;   // 8192

    float* ws       = (float*)d_ws;
    float* edge     = ws;                                  // N
    float* maxprob  = ws + (size_t)N;                      // N
    float* quality  = ws + (size_t)2 * N;                  // N
    unsigned* amaxU = (unsigned*)(ws + (size_t)3 * N);     // 1
    int* order      = (int*)(ws + (size_t)4 * N);          // N
    float* sboxes   = ws + (size_t)5 * N;                  // 4N
    unsigned* supmat = (unsigned*)(ws + (size_t)10 * N);   // N*NW words (8 MB)

    float* outMain = (float*)d_out;                // N*6
    float* keepf   = outMain + (size_t)6 * N;      // N
    float* labout  = outMain + (size_t)7 * N;      // N

    const int TB = 256;
    const int GB = (N + TB - 1) / TB;

    init_kernel<<<GB, TB, 0, stream>>>(keepf, amaxU, N);
    edge_kernel<<<N, 256, 0, stream>>>(feat, edge, N);
    maxprob_kernel<<<N, 256, 0, stream>>>(logits, maxprob, N, C);
    area_kernel<<<GB, TB, 0, stream>>>(boxes, amaxU, N);
    quality_kernel<<<GB, TB, 0, stream>>>(boxes, scores, edge, maxprob, amaxU, quality, N);
    sort_kernel<<<1, 1024, (size_t)NP * 8, stream>>>(scores, quality, order, N, NP);
    gather_kernel<<<GB, TB, 0, stream>>>(boxes, order, sboxes, N);
    iou_mask_kernel<<<N, NW, 0, stream>>>(sboxes, supmat, N, NW);
    nms_scan_kernel<<<1, NW, 0, stream>>>(quality, order, supmat, keepf, N, NW);
    finalize_kernel<<<GB, TB, 0, stream>>>(boxes, scores, quality, labels, keepf, outMain, labout, N);
}